// PointTransformerBlock_23338852286545
// MI455X (gfx1250) — compile-verified
//
#include <hip/hip_runtime.h>
#include <math.h>

// Problem constants (from reference)
#define NPTS   32768
#define BGRAPH 4
#define NPER   8192      // points per graph
#define MPER   2048      // sampled per graph
#define MTOT   8192      // B * MPER
#define CIN    64
#define CH     128
#define KNN    16
#define R2     0.01f

typedef __attribute__((ext_vector_type(2))) float v2f;
typedef __attribute__((ext_vector_type(8))) float v8f;

// ---------------------------------------------------------------------------
// fp32 WMMA 16x16x4 wrapper (D = A*B + C).  A: 16x4, B: 4x16, C/D: 16x16.
// Layout per CDNA5 ISA 7.12.2:
//   A: lanes0-15 rows M=0..15 {K=k,k+1}, lanes16-31 rows M=0..15 {K=k+2,k+3}
//   B: (mirror) lanes hold N=0..15; vgpr0={K=k | K=k+2}, vgpr1={K=k+1 | K=k+3}
//   C/D: vgpr v = row (v + 8*half), col = lane&15
// ---------------------------------------------------------------------------
__device__ __forceinline__ v8f wmma4(v2f a, v2f b, v8f c) {
  return __builtin_amdgcn_wmma_f32_16x16x4_f32(false, a, false, b, (short)0, c,
                                               false, false);
}

// 4 row-tiles (rows 0/16/32/48) x one 16-col tile of A[64 x 128] * W[128 x 128].
// B fragment is loaded once per k-step and reused across the 4 row tiles.
// Unroll 8 k-steps so the scheduler can cluster loads ahead of the WMMA chain.
template <bool NEGB>
__device__ __forceinline__ void gemm_col4(const float* __restrict__ A,
                                          const float* __restrict__ W,
                                          int col0, v8f acc[4]) {
  const int lane = threadIdx.x & 31;
  const int half = lane >> 4;
  const int r = lane & 15;
#pragma unroll 8
  for (int k = 0; k < CH; k += 4) {
    v2f b;
    float b0 = W[(k + 2 * half) * CH + col0 + r];
    float b1 = W[(k + 2 * half + 1) * CH + col0 + r];
    if (NEGB) { b0 = -b0; b1 = -b1; }
    b.x = b0; b.y = b1;
#pragma unroll
    for (int rt = 0; rt < 4; ++rt) {
      v2f a;
      a.x = A[(rt * 16 + r) * CH + k + 2 * half];
      a.y = A[(rt * 16 + r) * CH + k + 2 * half + 1];
      acc[rt] = wmma4(a, b, acc[rt]);
    }
  }
}

// Async copy: one 512B row per wave-instruction (32 lanes x 16B), LDS dest.
// GLOBAL_LOAD_ASYNC_TO_LDS_B128, GV mode (SADDR = off), tracked by ASYNCcnt.
__device__ __forceinline__ void async_row_to_lds(unsigned lds_off,
                                                 const float* gsrc) {
  unsigned long long ga = (unsigned long long)(uintptr_t)gsrc;
  asm volatile("global_load_async_to_lds_b128 %0, %1, off" ::"v"(lds_off),
               "v"(ga)
               : "memory");
}

// ---------------------------------------------------------------------------
// K1: xc = x @ W_in + b_in   [32768,64]x[64,128] -> [32768,128]
// 256 threads = 8 waves; each wave does a 16-row strip x 128 cols.
// ---------------------------------------------------------------------------
__global__ __launch_bounds__(256) void k_inproj(const float* __restrict__ x,
                                                const float* __restrict__ Win,
                                                const float* __restrict__ bin,
                                                float* __restrict__ xc) {
  const int wave = threadIdx.x >> 5;
  const int lane = threadIdx.x & 31;
  const int half = lane >> 4;
  const int r = lane & 15;
  const int row0 = blockIdx.x * 128 + wave * 16;

  v8f acc[8];
#pragma unroll
  for (int t = 0; t < 8; ++t) {
    float bv = bin[t * 16 + r];
#pragma unroll
    for (int v = 0; v < 8; ++v) acc[t][v] = bv;
  }
  const float* arow = x + (row0 + r) * CIN;
  for (int k = 0; k < CIN; k += 4) {
    v2f a;
    a.x = arow[k + 2 * half];
    a.y = arow[k + 2 * half + 1];
#pragma unroll
    for (int t = 0; t < 8; ++t) {
      v2f b;
      b.x = Win[(k + 2 * half) * CH + t * 16 + r];
      b.y = Win[(k + 2 * half + 1) * CH + t * 16 + r];
      acc[t] = wmma4(a, b, acc[t]);
    }
  }
#pragma unroll
  for (int t = 0; t < 8; ++t)
#pragma unroll
    for (int v = 0; v < 8; ++v)
      xc[(row0 + v + 8 * half) * CH + t * 16 + r] = acc[t][v];
}

// ---------------------------------------------------------------------------
// K2: pnorm[i] = |pos_i|^2
// ---------------------------------------------------------------------------
__global__ __launch_bounds__(256) void k_pnorm(const float* __restrict__ pos,
                                               float* __restrict__ pnorm) {
  int i = blockIdx.x * 256 + threadIdx.x;
  if (i < NPTS) {
    float a = pos[3 * i], b = pos[3 * i + 1], c = pos[3 * i + 2];
    pnorm[i] = a * a + b * b + c * c;
  }
}

// ---------------------------------------------------------------------------
// K3: farthest point sampling, one 1024-thread block per graph.
// mind lives in registers (8 pts/thread). Per step: local update + argmax,
// shfl wave-reduce, 32-entry LDS reduce. Also emits pos_q and batch_out.
// ---------------------------------------------------------------------------
__global__ __launch_bounds__(1024) void k_fps(const float* __restrict__ pos,
                                              int* __restrict__ idx,
                                              float* __restrict__ posq,
                                              int* __restrict__ batch_out) {
  const int b = blockIdx.x;
  const int t = threadIdx.x;
  const float* P = pos + (size_t)b * NPER * 3;

  float px[8], py[8], pz[8], mind[8];
#pragma unroll
  for (int j = 0; j < 8; ++j) {
    int p = t * 8 + j;
    px[j] = P[3 * p];
    py[j] = P[3 * p + 1];
    pz[j] = P[3 * p + 2];
    mind[j] = 1e10f;
  }
  __shared__ float s_val[32];
  __shared__ int s_idx[32];
  __shared__ int s_last;

  int last = 0;
  if (t == 0) idx[b * MPER] = 0;

  for (int s = 1; s < MPER; ++s) {
    float lx = P[3 * last], ly = P[3 * last + 1], lz = P[3 * last + 2];
    float bestv = -1.0f;
    int besti = 0x7fffffff;
#pragma unroll
    for (int j = 0; j < 8; ++j) {
      float dx = px[j] - lx, dy = py[j] - ly, dz = pz[j] - lz;
      float d = dx * dx + dy * dy + dz * dz;
      mind[j] = fminf(mind[j], d);
      if (mind[j] > bestv) { bestv = mind[j]; besti = t * 8 + j; }
    }
    // wave32 max-reduce, tie -> lowest index (matches jnp.argmax)
#pragma unroll
    for (int off = 16; off; off >>= 1) {
      float ov = __shfl_xor(bestv, off, 32);
      int oi = __shfl_xor(besti, off, 32);
      if (ov > bestv || (ov == bestv && oi < besti)) { bestv = ov; besti = oi; }
    }
    if ((t & 31) == 0) { s_val[t >> 5] = bestv; s_idx[t >> 5] = besti; }
    __syncthreads();
    if (t < 32) {
      bestv = s_val[t];
      besti = s_idx[t];
#pragma unroll
      for (int off = 16; off; off >>= 1) {
        float ov = __shfl_xor(bestv, off, 32);
        int oi = __shfl_xor(besti, off, 32);
        if (ov > bestv || (ov == bestv && oi < besti)) { bestv = ov; besti = oi; }
      }
      if (t == 0) { s_last = besti; idx[b * MPER + s] = besti; }
    }
    __syncthreads();
    last = s_last;
  }
  // pos_q gather + batch_out
  for (int q = t; q < MPER; q += 1024) {
    int li = idx[b * MPER + q];
    posq[(b * MPER + q) * 3 + 0] = P[3 * li];
    posq[(b * MPER + q) * 3 + 1] = P[3 * li + 1];
    posq[(b * MPER + q) * 3 + 2] = P[3 * li + 2];
    batch_out[b * MPER + q] = b;
  }
}

// ---------------------------------------------------------------------------
// K4: radius-capped 16-NN. One wave per query (8 queries / 256-thread block).
// d2 = |q|^2 + |p|^2 - 2 q.p (matches reference expansion). Per-lane sorted
// top-16 in registers, then 16 selection rounds over 512 LDS candidates.
// ---------------------------------------------------------------------------
__global__ __launch_bounds__(256) void k_knn(const float* __restrict__ pos,
                                             const float* __restrict__ pnorm,
                                             const int* __restrict__ idx,
                                             int* __restrict__ nbr,
                                             float* __restrict__ nd2) {
  __shared__ float cd[8][512];
  __shared__ int ci[8][512];
  const int wave = threadIdx.x >> 5;
  const int lane = threadIdx.x & 31;
  const int gq = blockIdx.x * 8 + wave;
  const int b = gq >> 11;  // / MPER
  const int qi = idx[gq];
  const float* P = pos + (size_t)b * NPER * 3;
  const float* PN = pnorm + (size_t)b * NPER;
  const float qx = P[3 * qi], qy = P[3 * qi + 1], qz = P[3 * qi + 2];
  const float qn = qx * qx + qy * qy + qz * qz;

  float bd[16];
  int bi[16];
#pragma unroll
  for (int u = 0; u < 16; ++u) { bd[u] = 3e38f; bi[u] = 0x7fffffff; }

  for (int j = lane; j < NPER; j += 32) {
    float d = qn + PN[j] -
              2.0f * (qx * P[3 * j] + qy * P[3 * j + 1] + qz * P[3 * j + 2]);
    int id = j;
#pragma unroll
    for (int u = 0; u < 16; ++u) {  // sorted insert, tie -> lower index
      bool lt = (d < bd[u]) || (d == bd[u] && id < bi[u]);
      float od = bd[u]; int oi = bi[u];
      if (lt) { bd[u] = d; bi[u] = id; d = od; id = oi; }
    }
  }
#pragma unroll
  for (int u = 0; u < 16; ++u) {
    cd[wave][lane * 16 + u] = bd[u];
    ci[wave][lane * 16 + u] = bi[u];
  }
  __syncthreads();
  for (int round = 0; round < 16; ++round) {
    float mv = 3e38f;
    int mi = 0x7fffffff, mp = 0;
#pragma unroll
    for (int u = 0; u < 16; ++u) {
      float v = cd[wave][lane * 16 + u];
      int vi = ci[wave][lane * 16 + u];
      if (v < mv || (v == mv && vi < mi)) { mv = v; mi = vi; mp = lane * 16 + u; }
    }
#pragma unroll
    for (int off = 16; off; off >>= 1) {
      float ov = __shfl_xor(mv, off, 32);
      int oi = __shfl_xor(mi, off, 32);
      int op = __shfl_xor(mp, off, 32);
      if (ov < mv || (ov == mv && oi < mi)) { mv = ov; mi = oi; mp = op; }
    }
    if (lane == (mp >> 4)) cd[wave][mp] = 3e38f;  // owning lane invalidates
    if (lane == 0) { nbr[gq * 16 + round] = mi; nd2[gq * 16 + round] = mv; }
    __syncthreads();
  }
}

// ---------------------------------------------------------------------------
// K5: xqd = x_q @ W_d  (gathered rows of xc), [8192,128]x[128,128]
// ---------------------------------------------------------------------------
__global__ __launch_bounds__(256) void k_xqd(const float* __restrict__ xc,
                                             const int* __restrict__ idx,
                                             const float* __restrict__ Wd,
                                             float* __restrict__ xqd) {
  const int wave = threadIdx.x >> 5;
  const int lane = threadIdx.x & 31;
  const int half = lane >> 4;
  const int r = lane & 15;
  const int row0 = blockIdx.x * 128 + wave * 16;
  const int gq = row0 + r;
  const int b = gq >> 11;
  const float* arow = xc + (size_t)(b * NPER + idx[gq]) * CH;

  v8f acc[8];
#pragma unroll
  for (int t = 0; t < 8; ++t)
#pragma unroll
    for (int v = 0; v < 8; ++v) acc[t][v] = 0.0f;

#pragma unroll 8
  for (int k = 0; k < CH; k += 4) {
    v2f a;
    a.x = arow[k + 2 * half];
    a.y = arow[k + 2 * half + 1];
#pragma unroll
    for (int t = 0; t < 8; ++t) {
      v2f bb;
      bb.x = Wd[(k + 2 * half) * CH + t * 16 + r];
      bb.y = Wd[(k + 2 * half + 1) * CH + t * 16 + r];
      acc[t] = wmma4(a, bb, acc[t]);
    }
  }
#pragma unroll
  for (int t = 0; t < 8; ++t)
#pragma unroll
    for (int v = 0; v < 8; ++v)
      xqd[(row0 + v + 8 * half) * CH + t * 16 + r] = acc[t][v];
}

// ---------------------------------------------------------------------------
// K6: fused message / attention / aggregate / LayerNorm.
// 4 queries (64 neighbor-rows) per 256-thread block, 8 waves.
// Each wave owns ONE 16-col tile across all 4 row tiles (B fragment reuse x4).
// x_j gather uses GLOBAL_LOAD_ASYNC_TO_LDS_B128 (ASYNCcnt).
// LDS: bufX(x_j->alpha) | bufH(h1->vd) | bufD(delta->alpha_pre) | obuf | ndbuf
// ---------------------------------------------------------------------------
__global__ __launch_bounds__(256) void k_msg(
    const float* __restrict__ xc, const float* __restrict__ pos,
    const float* __restrict__ posq, const int* __restrict__ nbr,
    const float* __restrict__ nd2, const float* __restrict__ xqd,
    const float* __restrict__ Wp1, const float* __restrict__ bp1,
    const float* __restrict__ Wp2, const float* __restrict__ bp2,
    const float* __restrict__ Wa, const float* __restrict__ ba,
    const float* __restrict__ Wv, const float* __restrict__ Ws,
    const float* __restrict__ gamma, const float* __restrict__ beta,
    float* __restrict__ y) {
  extern __shared__ float smem[];
  float* bufX = smem;           // 64*128  x_j, later alpha
  float* bufH = smem + 8192;    // 64*128  h1, later vd = x_j@Wv + delta
  float* bufD = smem + 16384;   // 64*128  delta, later alpha_pre
  float* obuf = smem + 24576;   // 512     rel (tmp), later out pre-LN
  float* ndbuf = smem + 25088;  // 64      nd2 for this block's rows

  const int q0 = blockIdx.x * 4;
  const int wave = threadIdx.x >> 5;
  const int lane = threadIdx.x & 31;
  const int half = lane >> 4;
  const int r = lane & 15;
  const int col0 = wave * 16;  // this wave's column tile

  // Warm GL2/WGP$ for the shared weight matrices (global_prefetch).
  {
    int o = threadIdx.x * 64;  // covers the 64KB matrix at 256B spacing
    __builtin_prefetch(Wp2 + o, 0, 0);
    __builtin_prefetch(Wv + o, 0, 0);
    __builtin_prefetch(Ws + o, 0, 0);
    __builtin_prefetch(Wa + o, 0, 0);
  }

  // ---- phase 0: async-gather x_j rows into LDS; rel, nd2 cache, h1 ----
#pragma unroll
  for (int i = 0; i < 8; ++i) {
    int row = wave * 8 + i;
    int gq = q0 + (row >> 4);
    int j = nbr[gq * 16 + (row & 15)];
    int b = gq >> 11;
    const float* src = xc + (size_t)(b * NPER + j) * CH + lane * 4;
    unsigned lds_off =
        (unsigned)((const char*)(bufX + row * CH) - (const char*)smem) +
        (unsigned)(lane * 16);
    async_row_to_lds(lds_off, src);
  }
  if (threadIdx.x < 64) {
    int row = threadIdx.x;
    int gq = q0 + (row >> 4);
    int j = nbr[gq * 16 + (row & 15)];
    int b = gq >> 11;
    const float* pj = pos + (size_t)(b * NPER + j) * 3;
    const float* pq = posq + (size_t)gq * 3;
    obuf[row * 3 + 0] = pq[0] - pj[0];
    obuf[row * 3 + 1] = pq[1] - pj[1];
    obuf[row * 3 + 2] = pq[2] - pj[2];
    ndbuf[row] = nd2[q0 * 16 + row];  // rows are contiguous: q0*16 .. +63
  }
  asm volatile("s_wait_asynccnt 0x0" ::: "memory");
  __syncthreads();
  for (int e = threadIdx.x; e < 64 * CH; e += 256) {
    int row = e >> 7, c = e & 127;
    float h = obuf[row * 3] * Wp1[c] + obuf[row * 3 + 1] * Wp1[CH + c] +
              obuf[row * 3 + 2] * Wp1[2 * CH + c] + bp1[c];
    bufH[e] = fmaxf(h, 0.0f);
  }
  __syncthreads();

  // ---- phase 1: delta = relu(h1 @ Wp2 + bp2) -> bufD ----
  {
    v8f acc[4];
    float bv = bp2[col0 + r];
#pragma unroll
    for (int rt = 0; rt < 4; ++rt)
#pragma unroll
      for (int v = 0; v < 8; ++v) acc[rt][v] = bv;
    gemm_col4<false>(bufH, Wp2, col0, acc);
#pragma unroll
    for (int rt = 0; rt < 4; ++rt)
#pragma unroll
      for (int v = 0; v < 8; ++v)
        bufD[(rt * 16 + v + 8 * half) * CH + col0 + r] = fmaxf(acc[rt][v], 0.0f);
  }
  __syncthreads();

  // ---- phase 2: vd = x_j @ Wv + delta -> bufH ----
  {
    v8f acc[4];
#pragma unroll
    for (int rt = 0; rt < 4; ++rt)
#pragma unroll
      for (int v = 0; v < 8; ++v)
        acc[rt][v] = bufD[(rt * 16 + v + 8 * half) * CH + col0 + r];
    gemm_col4<false>(bufX, Wv, col0, acc);
#pragma unroll
    for (int rt = 0; rt < 4; ++rt)
#pragma unroll
      for (int v = 0; v < 8; ++v)
        bufH[(rt * 16 + v + 8 * half) * CH + col0 + r] = acc[rt][v];
  }
  __syncthreads();

  // ---- phase 3: alpha_pre = xqd - x_j@Ws + delta  (neg B on load) -> bufD --
  {
    v8f acc[4];
#pragma unroll
    for (int rt = 0; rt < 4; ++rt)
#pragma unroll
      for (int v = 0; v < 8; ++v) {
        int row = rt * 16 + v + 8 * half;
        int gq = q0 + (row >> 4);
        acc[rt][v] = bufD[row * CH + col0 + r] + xqd[gq * CH + col0 + r];
      }
    gemm_col4<true>(bufX, Ws, col0, acc);
#pragma unroll
    for (int rt = 0; rt < 4; ++rt)
#pragma unroll
      for (int v = 0; v < 8; ++v)
        bufD[(rt * 16 + v + 8 * half) * CH + col0 + r] = acc[rt][v];
  }
  __syncthreads();

  // ---- phase 4: alpha = relu(alpha_pre @ Wa + ba) -> bufX ----
  {
    v8f acc[4];
    float bv = ba[col0 + r];
#pragma unroll
    for (int rt = 0; rt < 4; ++rt)
#pragma unroll
      for (int v = 0; v < 8; ++v) acc[rt][v] = bv;
    gemm_col4<false>(bufD, Wa, col0, acc);
#pragma unroll
    for (int rt = 0; rt < 4; ++rt)
#pragma unroll
      for (int v = 0; v < 8; ++v)
        bufX[(rt * 16 + v + 8 * half) * CH + col0 + r] = fmaxf(acc[rt][v], 0.0f);
  }
  __syncthreads();

  // ---- phase 5: masked softmax over K and weighted sum of vd ----
  for (int e = threadIdx.x; e < 4 * CH; e += 256) {
    int q = e >> 7, c = e & 127;
    float av[16];
    float mx = -3e38f;
#pragma unroll
    for (int k = 0; k < 16; ++k) {
      float a = bufX[(q * 16 + k) * CH + c];
      a = (ndbuf[q * 16 + k] <= R2) ? a : -1e9f;
      av[k] = a;
      mx = fmaxf(mx, a);
    }
    float s = 0.0f, o = 0.0f;
#pragma unroll
    for (int k = 0; k < 16; ++k) {
      float ea = __expf(av[k] - mx);
      s += ea;
      o += ea * bufH[(q * 16 + k) * CH + c];
    }
    obuf[q * CH + c] = o / s;
  }
  __syncthreads();

  // ---- phase 6: LayerNorm per query (wave w < 4 handles query w) ----
  if (wave < 4) {
    int q = wave, gq = q0 + q;
    float v[4], s = 0.0f;
#pragma unroll
    for (int u = 0; u < 4; ++u) {
      v[u] = obuf[q * CH + lane + 32 * u];
      s += v[u];
    }
#pragma unroll
    for (int off = 16; off; off >>= 1) s += __shfl_xor(s, off, 32);
    float mu = s * (1.0f / CH);
    float ss = 0.0f;
#pragma unroll
    for (int u = 0; u < 4; ++u) {
      float d = v[u] - mu;
      ss += d * d;
    }
#pragma unroll
    for (int off = 16; off; off >>= 1) ss += __shfl_xor(ss, off, 32);
    float rs = rsqrtf(ss * (1.0f / CH) + 1e-5f);
#pragma unroll
    for (int u = 0; u < 4; ++u) {
      int c = lane + 32 * u;
      y[gq * CH + c] = (v[u] - mu) * rs * gamma[c] + beta[c];
    }
  }
}

// ---------------------------------------------------------------------------
// Launch
// ---------------------------------------------------------------------------
extern "C" void kernel_launch(void* const* d_in, const int* in_sizes, int n_in,
                              void* d_out, int out_size, void* d_ws,
                              size_t ws_size, hipStream_t stream) {
  const float* x    = (const float*)d_in[0];
  const float* pos  = (const float*)d_in[1];
  // d_in[2] = batch (layout known; unused)
  const float* Win  = (const float*)d_in[3];
  const float* bin  = (const float*)d_in[4];
  const float* Wp1  = (const float*)d_in[5];
  const float* bp1  = (const float*)d_in[6];
  const float* Wp2  = (const float*)d_in[7];
  const float* bp2  = (const float*)d_in[8];
  const float* Wa   = (const float*)d_in[9];
  const float* ba   = (const float*)d_in[10];
  const float* Wv   = (const float*)d_in[11];
  const float* Ws   = (const float*)d_in[12];
  const float* Wd   = (const float*)d_in[13];
  const float* gamma = (const float*)d_in[14];
  const float* beta  = (const float*)d_in[15];

  float* y = (float*)d_out;                 // [8192,128]
  float* posq = y + (size_t)MTOT * CH;      // [8192,3]
  int* batch_out = (int*)(posq + (size_t)MTOT * 3);  // [8192] int32 bits

  float* ws = (float*)d_ws;
  float* xc    = ws;                          // 32768*128
  float* xqd   = xc + (size_t)NPTS * CH;      // 8192*128
  float* pnorm = xqd + (size_t)MTOT * CH;     // 32768
  float* nd2   = pnorm + NPTS;                // 8192*16
  int* idx = (int*)(nd2 + (size_t)MTOT * KNN);  // 8192
  int* nbr = idx + MTOT;                        // 8192*16

  k_inproj<<<NPTS / 128, 256, 0, stream>>>(x, Win, bin, xc);
  k_pnorm<<<NPTS / 256, 256, 0, stream>>>(pos, pnorm);
  k_fps<<<BGRAPH, 1024, 0, stream>>>(pos, idx, posq, batch_out);
  k_knn<<<MTOT / 8, 256, 0, stream>>>(pos, pnorm, idx, nbr, nd2);
  k_xqd<<<MTOT / 128, 256, 0, stream>>>(xc, idx, Wd, xqd);

  const size_t smem_bytes = (3 * 64 * CH + 512 + 64) * sizeof(float);
  k_msg<<<MTOT / 4, 256, smem_bytes, stream>>>(xc, pos, posq, nbr, nd2, xqd,
                                               Wp1, bp1, Wp2, bp2, Wa, ba, Wv,
                                               Ws, gamma, beta, y);
}